// Recurrent_FullAttention_88751204204837
// MI455X (gfx1250) — compile-verified
//
#include <hip/hip_runtime.h>

// Problem constants (from reference): B=16, L=256, E=32, NF=7, R=224
#define BATCH 16
#define LSEQ  256
#define EDIM  32
#define NFEAT 7
#define RDIM  (EDIM * NFEAT)   // 224
#define MMAX  (LSEQ * NFEAT)   // 1792 (max rows of the state matrix)
#define NT    512              // 16 wave32 waves per workgroup
#define NWAVES (NT / 32)

// Manual LDS arena layout (floats). sc FIRST so its accesses get small
// immediate DS offsets; st (229 KB) last.
#define SC_OFF     0
#define SC_SZ      (MMAX * 8 + 8)        // 14344
#define OUTB_OFF   (SC_OFF + SC_SZ)      // 14344
#define OUTB_SZ    (EDIM * 16)           // 512
#define QB_OFF     (OUTB_OFF + OUTB_SZ)  // 14856
#define QB_SZ      (EDIM * 8 + 16)       // 272
#define K0_OFF     (QB_OFF + QB_SZ)      // 15128
#define K0_SZ      (EDIM * 8)            // 256
#define RED_OFF    (K0_OFF + K0_SZ)      // 15384
#define RED_SZ     (8 * 64)              // 512
#define CMAX_OFF   (RED_OFF + RED_SZ)    // 15896
#define CSUM_OFF   (CMAX_OFF + 8)        // 15904
#define ST_OFF     (CSUM_OFF + 8)        // 15912 (byte 63648, 32B aligned)
#define ST_SZ      (MMAX * EDIM)         // 57344
#define ARENA      (ST_OFF + ST_SZ)      // 73256 floats = 293024 B

typedef float v2f __attribute__((ext_vector_type(2)));
typedef float v8f __attribute__((ext_vector_type(8)));

__global__ __launch_bounds__(NT, 1)
void recurrent_attn_kernel(const float* __restrict__ q,
                           const float* __restrict__ k,
                           const float* __restrict__ v,
                           float* __restrict__ out) {
  __shared__ __align__(16) float smem[ARENA];
  float* const sc     = smem + SC_OFF;    // scores -> exp(scores), [sp][r] stride 8
  float* const outbuf = smem + OUTB_OFF;  // partial v_t accumulator 32x16
  float* const qbuf   = smem + QB_OFF;    // Q_t[e][r] (+pad)
  float* const k0buf  = smem + K0_OFF;    // K_0[e][p] (t=0 only)
  float* const red    = smem + RED_OFF;   // softmax reduction scratch
  float* const colmax = smem + CMAX_OFF;
  float* const colsum = smem + CSUM_OFF;
  float* const st     = smem + ST_OFF;    // state KM[sp][e], sp = s*7+p

  const int b     = blockIdx.x;
  const int tid   = threadIdx.x;
  const int lane  = tid & 31;
  const int wave  = tid >> 5;
  const int m16   = lane & 15;   // row-in-tile / column index
  const int khalf = lane >> 4;   // selects K pair {0,1} vs {2,3}

  const size_t bbase = (size_t)b * LSEQ * RDIM;

  // ---- Load initial state: values (s,e,p) -> LDS (sp=s*7+p, e) ----
  for (int idx = tid; idx < LSEQ * RDIM; idx += NT) {
    int p  = idx % NFEAT;
    int se = idx / NFEAT;
    int e  = se % EDIM;
    int s  = se / EDIM;
    st[(s * NFEAT + p) * EDIM + e] = v[bbase + idx];
  }
  for (int idx = tid; idx < RDIM; idx += NT) {
    int e = idx / NFEAT, r = idx % NFEAT;
    qbuf[e * 8 + r]  = q[bbase + idx];
    k0buf[e * 8 + r] = k[bbase + idx];
  }
  __syncthreads();

  // ---- step t = 0: s0(7x7) = K0^T Q0 over e; softmax over p; v0 = V0 * a0 ----
  if (tid < NFEAT * NFEAT) {
    int p = tid / NFEAT, r = tid % NFEAT;
    float s0 = 0.f;
    for (int e = 0; e < EDIM; ++e)
      s0 += qbuf[e * 8 + r] * k0buf[e * 8 + p];
    sc[p * 8 + r] = s0;
  }
  __syncthreads();
  if (tid < NFEAT) {
    int r = tid;
    float mx = -3.402823466e+38f;
    for (int p = 0; p < NFEAT; ++p) mx = fmaxf(mx, sc[p * 8 + r]);
    float sum = 0.f;
    for (int p = 0; p < NFEAT; ++p) {
      float e0 = __expf(sc[p * 8 + r] - mx);
      sc[p * 8 + r] = e0;
      sum += e0;
    }
    float inv = 1.f / sum;
    for (int p = 0; p < NFEAT; ++p) sc[p * 8 + r] *= inv;
  }
  __syncthreads();
  float v0val = 0.f;
  if (tid < RDIM) {
    int e = tid / NFEAT, r = tid % NFEAT;
    for (int p = 0; p < NFEAT; ++p)
      v0val += st[p * EDIM + e] * sc[p * 8 + r];   // row s=0 still holds V0
  }
  __syncthreads();
  if (tid < RDIM) {
    int e = tid / NFEAT, r = tid % NFEAT;
    st[r * EDIM + e] = v0val;          // state[:,0] = v0
    out[bbase + tid] = v0val;          // out[b,0,e,r]
  }

  // ---- steps t = 1 .. 255 ----
  for (int t = 1; t < LSEQ; ++t) {
    const int M    = (t + 1) * NFEAT;        // valid rows (s <= t)
    const int Mpad = (M + 15) & ~15;         // <= 1792
    const size_t qrow = bbase + (size_t)t * RDIM;

    __syncthreads();  // st row t-1 complete; sc free for reuse
    if (tid < RDIM) {
      int e = tid / NFEAT, r = tid % NFEAT;
      qbuf[e * 8 + r] = q[qrow + tid];
      if (t + 1 < LSEQ) __builtin_prefetch(&q[qrow + RDIM + tid], 0, 1);
    }
    if (tid < EDIM * 16) outbuf[tid] = 0.f;
    __syncthreads();

    // B fragments of Q for the 8 K-chunks (K dim = e), unguarded:
    // lanes with n>=7 read adjacent data -> feed only discarded D columns.
    // B(4x16): vgpr v holds K = 2*khalf + v, N = lane&15.
    v2f bq[8];
    #pragma unroll
    for (int ke = 0; ke < 8; ++ke) {
      int e0 = ke * 4 + 2 * khalf;
      bq[ke].x = qbuf[e0 * 8 + m16];
      bq[ke].y = qbuf[e0 * 8 + 8 + m16];
    }

    // Phase B: scores = KM (Mx32) x Q (32x7), M-tiled across waves.
    for (int mt = wave; mt < (Mpad >> 4); mt += NWAVES) {
      v8f acc = {};
      int spA = mt * 16 + m16;  // A(16x4): lane row, vgpr v -> K = 2*khalf+v
      #pragma unroll
      for (int ke = 0; ke < 8; ++ke) {
        int e0 = ke * 4 + 2 * khalf;
        v2f a;
        a.x = st[spA * EDIM + e0];
        a.y = st[spA * EDIM + e0 + 1];
        acc = __builtin_amdgcn_wmma_f32_16x16x4_f32(
            false, a, false, bq[ke], (short)0, acc, false, false);
      }
      // D(16x16): vgpr v -> M = v + khalf*8, N = lane&15.
      // Store cols 0..7 unconditionally (col 7 is a dead column).
      if (m16 < 8) {
        int base = (mt * 16 + khalf * 8) * 8 + m16;
        #pragma unroll
        for (int vv = 0; vv < 8; ++vv)
          sc[base + vv * 8] = acc[vv];
      }
    }
    __syncthreads();

    // Phase C: masked column softmax over sp < M (7 columns).
    const int r     = tid & 7;
    const int chunk = tid >> 3;               // 0..63
    float lmax = -3.402823466e+38f;
    if (r < NFEAT)
      for (int sp = chunk; sp < M; sp += 64)
        lmax = fmaxf(lmax, sc[sp * 8 + r]);
    red[r * 64 + chunk] = lmax;
    __syncthreads();
    if (tid < 8) {
      float mx = -3.402823466e+38f;
      for (int c2 = 0; c2 < 64; ++c2) mx = fmaxf(mx, red[tid * 64 + c2]);
      colmax[tid] = mx;
    }
    __syncthreads();
    float lsum = 0.f;
    if (r < NFEAT) {
      float mx = colmax[r];
      for (int sp = chunk; sp < M; sp += 64) {
        float e0 = __expf(sc[sp * 8 + r] - mx);
        sc[sp * 8 + r] = e0;                 // unnormalized; divide at the end
        lsum += e0;
      }
    }
    // Zero the K-padding rows [M, Mpad) so phase D B-fragments need no guards.
    if (tid < (Mpad - M) * 8)
      sc[(M + (tid >> 3)) * 8 + (tid & 7)] = 0.f;
    red[r * 64 + chunk] = lsum;
    __syncthreads();
    if (tid < 8) {
      float sm = 0.f;
      for (int c2 = 0; c2 < 64; ++c2) sm += red[tid * 64 + c2];
      colsum[tid] = sm;
    }
    __syncthreads();

    // Phase D: v_t(32x7) = KM^T (32xM) x expS (Mx7); K(=sp) split across waves.
    // Fully unguarded fragment loads: rows >= M are zeroed, lanes n>=7 feed
    // only discarded output columns.
    const int et  = wave & 1;
    const int e0d = et * 16;
    v8f acc2 = {};
    for (int c = (wave >> 1); c < (Mpad >> 2); c += (NWAVES / 2)) {
      int spk0 = c * 4 + 2 * khalf;          // K for vgpr 0; +1 for vgpr 1
      v2f a, bb;
      a.x = st[spk0 * EDIM + e0d + m16];     // A[m][k] = KM^T[e0+m, sp0+k]
      a.y = st[(spk0 + 1) * EDIM + e0d + m16];
      bb.x = sc[spk0 * 8 + m16];
      bb.y = sc[spk0 * 8 + 8 + m16];
      acc2 = __builtin_amdgcn_wmma_f32_16x16x4_f32(
          false, a, false, bb, (short)0, acc2, false, false);
    }
    {
      #pragma unroll
      for (int vv = 0; vv < 8; ++vv) {
        int e = e0d + vv + khalf * 8;
        atomicAdd(&outbuf[e * 16 + m16], acc2[vv]);   // ds_add_f32
      }
    }
    __syncthreads();

    // Phase E: normalize, update state row t, write output row t.
    if (tid < RDIM) {
      int e = tid / NFEAT, p = tid % NFEAT;
      float val = outbuf[e * 16 + p] / colsum[p];
      st[(t * NFEAT + p) * EDIM + e] = val;
      out[qrow + tid] = val;                 // flat (e*7+p) == tid
    }
  }
}

extern "C" void kernel_launch(void* const* d_in, const int* in_sizes, int n_in,
                              void* d_out, int out_size, void* d_ws, size_t ws_size,
                              hipStream_t stream) {
  const float* q = (const float*)d_in[0];   // queries (16,256,224) f32
  const float* k = (const float*)d_in[1];   // keys
  const float* v = (const float*)d_in[2];   // values
  // d_in[3] = attn_mask (unused, mask_flag=False)
  float* out = (float*)d_out;               // (16,256,32,7) f32
  recurrent_attn_kernel<<<dim3(BATCH), dim3(NT), 0, stream>>>(q, k, v, out);
}